// Embedding_70171175682810
// MI455X (gfx1250) — compile-verified
//
#include <hip/hip_runtime.h>
#include <stdint.h>

// Embedding gather: out[t,h] = W[h, idx[t]] + b[h]
//   idx: [4096] int32, W: [128, 32000] f32 (row-major), b: [128] f32
//   out: [4096, 128] f32
//
// Pure data movement (no FLOPs) -> memory-roofline kernel:
//   ~16MB W footprint (fits L2), 2MB out  => ~1us at 23.3 TB/s.
// CDNA5 path exercised: async global->LDS copy (ASYNCcnt) for index staging.

#define V_DIM  32000
#define H_DIM  128
#define TOKENS 4096          // B*S = 16*256
#define WAVES_PER_BLOCK 8    // wave32: 256 threads/block
#define TOKENS_PER_BLOCK WAVES_PER_BLOCK

__global__ __launch_bounds__(256)
void embed_gather_kernel(const int* __restrict__ idx,
                         const float* __restrict__ W,
                         const float* __restrict__ bias,
                         float* __restrict__ out)
{
    __shared__ int s_idx[TOKENS_PER_BLOCK];

    const int tid       = threadIdx.x;
    const int wave      = tid >> 5;
    const int lane      = tid & 31;
    const int tokenBase = blockIdx.x * TOKENS_PER_BLOCK;

    // ---- CDNA5 async copy: stage this block's 8 token indices into LDS ----
    // Generic pointers into LDS carry the LDS byte offset in bits [31:0]
    // (ISA aperture rule), so truncation yields the DS address.
    if (tid < TOKENS_PER_BLOCK) {
        unsigned lds_addr = (unsigned)(uintptr_t)(&s_idx[tid]);
        uint64_t g_addr   = (uint64_t)(uintptr_t)(idx + tokenBase + tid);
        asm volatile("global_load_async_to_lds_b32 %0, %1, off"
                     :: "v"(lds_addr), "v"(g_addr)
                     : "memory");
    }
    asm volatile("s_wait_asynccnt 0" ::: "memory");  // waves with ASYNCcnt==0 pass through
    __syncthreads();

    // ---- one wave per token; lane owns h = 4*lane .. 4*lane+3 ----
    const int   token = tokenBase + wave;
    const int   v     = s_idx[wave];
    const float* col  = W + (size_t)v;       // element h is at col[h * V_DIM]
    const int   h0    = lane << 2;

    // 4 independent scattered loads (stride 128KB) -> all in flight at once.
    float4 r;
    r.x = col[(size_t)(h0 + 0) * V_DIM];
    r.y = col[(size_t)(h0 + 1) * V_DIM];
    r.z = col[(size_t)(h0 + 2) * V_DIM];
    r.w = col[(size_t)(h0 + 3) * V_DIM];

    // Coalesced 16B bias load (512B per wave, L0-resident across waves).
    const float4 bv = *reinterpret_cast<const float4*>(bias + h0);
    r.x += bv.x;  r.y += bv.y;  r.z += bv.z;  r.w += bv.w;

    // One aligned 16B store per lane: wave writes one contiguous 512B row.
    *reinterpret_cast<float4*>(out + (size_t)token * H_DIM + h0) = r;
}

extern "C" void kernel_launch(void* const* d_in, const int* in_sizes, int n_in,
                              void* d_out, int out_size, void* d_ws, size_t ws_size,
                              hipStream_t stream)
{
    (void)in_sizes; (void)n_in; (void)d_ws; (void)ws_size; (void)out_size;

    const int*   idx  = (const int*)d_in[0];    // src indices [B*S]
    const float* W    = (const float*)d_in[1];  // [H, V]
    const float* bias = (const float*)d_in[2];  // [H]
    float*       out  = (float*)d_out;          // [B*S, H]

    dim3 grid(TOKENS / TOKENS_PER_BLOCK);  // 512 blocks
    dim3 block(32 * WAVES_PER_BLOCK);      // 256 threads = 8 waves (wave32)
    hipLaunchKernelGGL(embed_gather_kernel, grid, block, 0, stream,
                       idx, W, bias, out);
}